// MultiBoxLoss_43877385895961
// MI455X (gfx1250) — compile-verified
//
#include <hip/hip_runtime.h>

typedef __attribute__((ext_vector_type(2))) float v2f;
typedef __attribute__((ext_vector_type(8))) float v8f;

#define NROWS            (512 * 256 * 4)       // 524288 rows per tensor
#define TILES_PER_TENSOR (NROWS / 16)          // 32768 tiles of 16 rows
#define NTILES           (3 * TILES_PER_TENSOR) // lt, lb, rb (rt is dropped by the reference)
#define NBLOCKS          1024
#define TPB              256

// Combined per-element weight: j==0 contributes unconditionally with weight 1
// (mask? w: d0^2 reduces to weight 1 either way); j in [1,54] gated by mask;
// j==55 is K-padding with weight 0.
__device__ __forceinline__ float feat_weight(int j, float mask) {
    if (j == 0)  return 1.0f;
    if (j < 3)   return 0.5f * mask;     // xy / 2
    if (j < 23)  return 0.05f * mask;    // q  / 20
    if (j < 55)  return 0.0625f * mask;  // lx,ly / 16
    return 0.0f;                         // pad feature
}

__global__ void __launch_bounds__(TPB)
multibox_loss_part1(const float* __restrict__ pred_lt,
                    const float* __restrict__ pred_lb,
                    const float* __restrict__ pred_rb,
                    const float* __restrict__ true_lt,
                    const float* __restrict__ true_lb,
                    const float* __restrict__ true_rb,
                    float* __restrict__ partials)
{
    const int lane       = threadIdx.x & 31;
    const int waveInBlk  = threadIdx.x >> 5;
    const int wavesPerBlk = TPB >> 5;
    const int waveId     = blockIdx.x * wavesPerBlk + waveInBlk;
    const int nWaves     = gridDim.x * wavesPerBlk;

    // WMMA f32 16x16x4 A-matrix layout: lanes 0-15 -> M=lane, K = {k0, k0+1}
    //                                   lanes16-31 -> M=lane-16, K = {k0+2, k0+3}
    const int m  = lane & 15;   // row within 16-row tile (fixed per lane)
    const int hi = lane >> 4;   // K-half select

    v2f bones; bones.x = 1.0f; bones.y = 1.0f;  // all-ones B -> D = row sums

    float acc = 0.0f;

    // NTILES (98304) is an exact multiple of nWaves (8192): uniform trip count,
    // EXEC stays all-1s for every WMMA.
    for (int tile = waveId; tile < NTILES; tile += nWaves) {
        int t  = tile / TILES_PER_TENSOR;
        int ti = tile - t * TILES_PER_TENSOR;
        const float* P; const float* T; float coef;
        if (t == 0)      { P = pred_lt; T = true_lt; coef = 2.0f; } // lt counted twice
        else if (t == 1) { P = pred_lb; T = true_lb; coef = 1.0f; }
        else             { P = pred_rb; T = true_rb; coef = 1.0f; }

        const size_t rowoff = (size_t)(ti * 16 + m) * 55u;
        const float* prow = P + rowoff;
        const float* trow = T + rowoff;

        // Speculative prefetch of this wave's next same-tensor tile
        // (dropped silently if OOB -> global_prefetch_b8).
        __builtin_prefetch(prow + (size_t)nWaves * 16u * 55u, 0, 0);
        __builtin_prefetch(trow + (size_t)nWaves * 16u * 55u, 0, 0);

        const float t0   = trow[0];
        const float mask = (truncf(t0) == 1.0f) ? 1.0f : 0.0f;

        v8f c = {};
#pragma unroll
        for (int ch = 0; ch < 14; ++ch) {        // K = 56 (55 features + 1 zero pad)
            const int j0 = ch * 4 + 2 * hi;
            const int j1 = j0 + 1;
            const int i0 = (j0 < 55) ? j0 : 54;  // clamp pad index (weight is 0)
            const int i1 = (j1 < 55) ? j1 : 54;
            const float d0 = prow[i0] - trow[i0];
            const float d1 = prow[i1] - trow[i1];
            v2f a;
            a.x = d0 * d0 * feat_weight(j0, mask);
            a.y = d1 * d1 * feat_weight(j1, mask);
            // D = A x 1 + C : accumulates weighted row sums (replicated over N)
            c = __builtin_amdgcn_wmma_f32_16x16x4_f32(
                    false, a, false, bones, (short)0, c, false, false);
        }

        float s = c[0] + c[1] + c[2] + c[3] + c[4] + c[5] + c[6] + c[7];
        acc += coef * s;   // every row counted 16x (16 identical columns); fixed later
    }

    // Deterministic wave32 butterfly reduction (ds-permute path).
#pragma unroll
    for (int off = 16; off >= 1; off >>= 1)
        acc += __shfl_xor(acc, off, 32);

    __shared__ float smem[TPB / 32];
    if (lane == 0) smem[waveInBlk] = acc;
    __syncthreads();
    if (threadIdx.x == 0) {
        float bsum = 0.0f;
#pragma unroll
        for (int i = 0; i < TPB / 32; ++i) bsum += smem[i];
        partials[blockIdx.x] = bsum;
    }
}

__global__ void __launch_bounds__(TPB)
multibox_loss_part2(const float* __restrict__ partials, float* __restrict__ out, int n)
{
    __shared__ float smem[TPB];
    float s = 0.0f;
    for (int i = threadIdx.x; i < n; i += TPB) s += partials[i];  // fixed order
    smem[threadIdx.x] = s;
    __syncthreads();
#pragma unroll
    for (int off = TPB / 2; off >= 1; off >>= 1) {
        if (threadIdx.x < off) smem[threadIdx.x] += smem[threadIdx.x + off];
        __syncthreads();
    }
    if (threadIdx.x == 0)
        out[0] = smem[0] * (1.0f / (16.0f * 512.0f));  // /16 column dup, /batch
}

extern "C" void kernel_launch(void* const* d_in, const int* in_sizes, int n_in,
                              void* d_out, int out_size, void* d_ws, size_t ws_size,
                              hipStream_t stream) {
    // setup_inputs order: pred_lt, pred_rt, pred_lb, pred_rb,
    //                     true_lt, true_rt, true_lb, true_rb  (rt pair unused)
    const float* pred_lt = (const float*)d_in[0];
    const float* pred_lb = (const float*)d_in[2];
    const float* pred_rb = (const float*)d_in[3];
    const float* true_lt = (const float*)d_in[4];
    const float* true_lb = (const float*)d_in[6];
    const float* true_rb = (const float*)d_in[7];
    float* partials = (float*)d_ws;           // NBLOCKS floats = 4 KB
    float* out      = (float*)d_out;

    multibox_loss_part1<<<NBLOCKS, TPB, 0, stream>>>(
        pred_lt, pred_lb, pred_rb, true_lt, true_lb, true_rb, partials);
    multibox_loss_part2<<<1, TPB, 0, stream>>>(partials, out, NBLOCKS);
}